// Interpolating_27041114096414
// MI455X (gfx1250) — compile-verified
//
#include <hip/hip_runtime.h>

typedef __attribute__((ext_vector_type(2))) float v2f;
typedef __attribute__((ext_vector_type(8))) float v8f;

#define BATCH 16
#define NT    4096
#define NS    1024
#define KNN   3
#define WAVES 4
#define TPB   (WAVES * 32)
#define NCTILE (NS / 16)      // 64 column tiles of 16 sampled points

__device__ __forceinline__ uint32_t umin32(uint32_t a, uint32_t b) { return a < b ? a : b; }
__device__ __forceinline__ uint32_t umax32(uint32_t a, uint32_t b) { return a > b ? a : b; }

// insert key into sorted (s0 <= s1 <= s2) keep-3-smallest, pure min/max network
__device__ __forceinline__ void ins3(uint32_t k, uint32_t& s0, uint32_t& s1, uint32_t& s2) {
    uint32_t lo0 = umin32(k, s0);
    uint32_t hi0 = umax32(k, s0);
    s0 = lo0;
    uint32_t lo1 = umin32(hi0, s1);
    uint32_t hi1 = umax32(hi0, s1);
    s1 = lo1;
    s2 = umin32(hi1, s2);
}

__global__ __launch_bounds__(TPB)
void Interpolating_27041114096414_kernel(const float* __restrict__ xu,
                                         const float* __restrict__ xs,
                                         float* __restrict__ out)
{
    // staged sampled points: [x, y, z, |b|^2]
    __shared__ float4   smp[NS];                      // 16 KB
    __shared__ uint32_t mkey[WAVES][16][48];          // 12 KB partial top-3 keys

    const int tid  = threadIdx.x;
    const int wave = tid >> 5;
    const int lane = tid & 31;
    const int n    = lane & 15;    // position within half-wave
    const int hi   = lane >> 4;    // half-wave select

    const int b       = blockIdx.x >> 6;                      // 64 blocks per batch
    const int tile    = (blockIdx.x & 63) * WAVES + wave;     // 16-row tile in batch
    const int rowBase = tile * 16;

    // ---- stage sampled points + norms into LDS (cooperative) ----
    const float* xsb = xs + (size_t)b * NS * 3;
    for (int p = tid; p < NS; p += TPB) {
        float x = xsb[p * 3 + 0];
        float y = xsb[p * 3 + 1];
        float z = xsb[p * 3 + 2];
        smp[p] = make_float4(x, y, z, x * x + y * y + z * z);
    }
    __syncthreads();

    // ---- build A fragment (16x4 f32): row = n ----
    // lanes 0-15 hold K0,K1 ; lanes 16-31 hold K2,K3  (per ISA 16x4 f32 A layout)
    // A row = [-2x, -2y, -2z, 1.0]
    const float* ap = xu + ((size_t)b * NT + rowBase + n) * 3;
    const float px = ap[0], py = ap[1], pz = ap[2];
    const float myNorm = px * px + py * py + pz * pz;
    v2f a;
    if (hi == 0) { a.x = -2.0f * px; a.y = -2.0f * py; }
    else         { a.x = -2.0f * pz; a.y = 1.0f;       }

    // per-lane broadcast of |a|^2 for the 8 D-rows this lane owns (rows hi*8 + r)
    float rnorm[8];
#pragma unroll
    for (int r = 0; r < 8; ++r)
        rnorm[r] = __shfl(myNorm, hi * 8 + r, 32);

    // running sorted top-3 keys per owned row:
    // key = (bits(max(d2,0)) & 0xFFFFFC00) | col   (order: d2 trunc, then col)
    uint32_t s0[8], s1[8], s2[8];
#pragma unroll
    for (int r = 0; r < 8; ++r) { s0[r] = s1[r] = s2[r] = 0xFFFFFFFFu; }

    // B fragment (4x16 f32): lanes 0-15 hold K0,K1 of col n; lanes 16-31 hold K2,K3
    const int boff = hi * 2;
    v2f bf;
    {
        const float* bp = (const float*)&smp[n];
        bf.x = bp[boff + 0];
        bf.y = bp[boff + 1];
    }

    // ---- sweep all sampled points in 16-column tiles via WMMA ----
    for (int ct = 0; ct < NCTILE; ++ct) {
        const uint32_t col = (uint32_t)(ct * 16 + n);

        // software-pipeline: fetch next tile's B fragment before consuming this one
        v2f bfn = bf;
        if (ct + 1 < NCTILE) {
            const float* bp = (const float*)&smp[col + 16];
            bfn.x = bp[boff + 0];
            bfn.y = bp[boff + 1];
        }

        v8f c = {};
        // D = A(16x4) * B(4x16) + 0  ->  D[row][col] = |b|^2 - 2 a.b
        c = __builtin_amdgcn_wmma_f32_16x16x4_f32(
                /*neg_a=*/false, a, /*neg_b=*/false, bf,
                /*c_mod=*/(short)0, c, /*reuse_a=*/false, /*reuse_b=*/false);

#pragma unroll
        for (int r = 0; r < 8; ++r) {
            const float d2 = fmaxf(c[r] + rnorm[r], 0.0f);   // + |a|^2, clamp >= 0
            const uint32_t k = (__float_as_uint(d2) & 0xFFFFFC00u) | col;
            ins3(k, s0[r], s1[r], s2[r]);                    // v_min_u32/v_max_u32 only
        }

        bf = bfn;
    }

    // ---- dump per-lane partial top-3 key lists to LDS ----
#pragma unroll
    for (int r = 0; r < 8; ++r) {
        const int row  = hi * 8 + r;
        const int base = n * 3;
        mkey[wave][row][base + 0] = s0[r];
        mkey[wave][row][base + 1] = s1[r];
        mkey[wave][row][base + 2] = s2[r];
    }
    __syncthreads();

    // ---- final merge: one lane per row scans 48 keys (min/max network) ----
    if (hi == 0) {
        uint32_t f0 = 0xFFFFFFFFu, f1 = 0xFFFFFFFFu, f2 = 0xFFFFFFFFu;
        const uint4* kp = (const uint4*)(&mkey[wave][n][0]);   // 16B-aligned (192B rows)
#pragma unroll
        for (int q = 0; q < 12; ++q) {
            const uint4 v = kp[q];
            ins3(v.x, f0, f1, f2);
            ins3(v.y, f0, f1, f2);
            ins3(v.z, f0, f1, f2);
            ins3(v.w, f0, f1, f2);
        }
        const int j0 = (int)(f0 & 1023u);
        const int j1 = (int)(f1 & 1023u);
        const int j2 = (int)(f2 & 1023u);

        // recompute EXACT distances for the winners (keys were truncated for ordering)
        const float4 q0 = smp[j0];
        const float4 q1 = smp[j1];
        const float4 q2 = smp[j2];
        const float e0 = (px - q0.x) * (px - q0.x) + (py - q0.y) * (py - q0.y) + (pz - q0.z) * (pz - q0.z);
        const float e1 = (px - q1.x) * (px - q1.x) + (py - q1.y) * (py - q1.y) + (pz - q1.z) * (pz - q1.z);
        const float e2 = (px - q2.x) * (px - q2.x) + (py - q2.y) * (py - q2.y) + (pz - q2.z) * (pz - q2.z);
        const float d0 = fmaxf(sqrtf(fmaxf(e0, 0.0f)), 1e-10f);
        const float d1 = fmaxf(sqrtf(fmaxf(e1, 0.0f)), 1e-10f);
        const float d2 = fmaxf(sqrtf(fmaxf(e2, 0.0f)), 1e-10f);
        float w0 = 1.0f / d0, w1 = 1.0f / d1, w2 = 1.0f / d2;
        const float inv = 1.0f / (w0 + w1 + w2);
        w0 *= inv; w1 *= inv; w2 *= inv;

        const size_t obase = ((size_t)b * NT + rowBase + n) * KNN;
        // output 0: indexes [B, NT, K] (written as float values into float buffer)
        out[obase + 0] = (float)j0;
        out[obase + 1] = (float)j1;
        out[obase + 2] = (float)j2;
        // output 1: weights [B, NT, K, 1], concatenated after indexes
        float* wout = out + (size_t)BATCH * NT * KNN;
        wout[obase + 0] = w0;
        wout[obase + 1] = w1;
        wout[obase + 2] = w2;
    }
}

extern "C" void kernel_launch(void* const* d_in, const int* in_sizes, int n_in,
                              void* d_out, int out_size, void* d_ws, size_t ws_size,
                              hipStream_t stream) {
    (void)in_sizes; (void)n_in; (void)out_size; (void)d_ws; (void)ws_size;
    const float* xu = (const float*)d_in[0];   // [16, 4096, 3]
    const float* xs = (const float*)d_in[1];   // [16, 1024, 3]
    float* out = (float*)d_out;                // [196608 idx-as-float | 196608 weights]

    // 64 blocks per batch * 16 batches; each block = 4 waves = 4 row-tiles of 16
    dim3 grid(BATCH * (NT / 16) / WAVES);
    dim3 block(TPB);
    Interpolating_27041114096414_kernel<<<grid, block, 0, stream>>>(xu, xs, out);
}